// Net_1503238553647
// MI455X (gfx1250) — compile-verified
//
#include <hip/hip_runtime.h>
#include <math.h>

// CDNA5 / gfx1250: wave32, WMMA f32 16x16x4
typedef float v2f __attribute__((ext_vector_type(2)));
typedef float v8f __attribute__((ext_vector_type(8)));

// ---------------- graph normalization ----------------

__global__ void k_deg(const int* __restrict__ ei, float* __restrict__ deg, int E) {
    int e = blockIdx.x * blockDim.x + threadIdx.x;
    if (e < E) atomicAdd(&deg[ei[e]], 1.0f);   // row = ei[0:E]
}

__global__ void k_dinv(const float* __restrict__ deg, float* __restrict__ dinv, int N) {
    int i = blockIdx.x * blockDim.x + threadIdx.x;
    if (i < N) {
        float d = deg[i];
        dinv[i] = (d > 0.0f) ? rsqrtf(fmaxf(d, 1e-30f)) : 0.0f;
    }
}

__global__ void k_w(const int* __restrict__ ei, const float* __restrict__ dinv,
                    float* __restrict__ w, int E) {
    int e = blockIdx.x * blockDim.x + threadIdx.x;
    if (e < E) w[e] = -dinv[ei[e]] * dinv[ei[E + e]];
}

// ---------------- weight pack: W[K x Nout] -> P[(K/2) x Np x 2], zero-padded cols ------
// P[((k>>1)*Np + n)*2 + (k&1)] = n < Nout ? W[k*Nout + n] : 0
// This lets each GEMM lane fetch its (k+kb, k+kb+1) B pair with ONE b64 load.

__global__ void k_pack_w(const float* __restrict__ W, float* __restrict__ P,
                         int K, int Nout, int Np) {
    int t = blockIdx.x * blockDim.x + threadIdx.x;
    if (t >= K * Np) return;
    int k = t / Np;
    int n = t - k * Np;
    float v = (n < Nout) ? W[(size_t)k * Nout + n] : 0.0f;
    P[(((size_t)(k >> 1)) * Np + n) * 2 + (k & 1)] = v;
}

// ---------------- SpMM: out[row] += w * X[col]  (atomic scatter, float4 chunks) --------

__global__ void k_spmm(const int* __restrict__ ei, const float* __restrict__ w,
                       const float* __restrict__ X, float* __restrict__ out,
                       int E, int chunks, int ld) {
    int t = blockIdx.x * blockDim.x + threadIdx.x;
    if (t >= E * chunks) return;
    int e = t / chunks;
    int c = (t - e * chunks) * 4;
    int r  = ei[e];
    int cl = ei[E + e];
    float we = w[e];
    const float4 v = *(const float4*)(X + (size_t)cl * ld + c);
    float* o = out + (size_t)r * ld + c;
    atomicAdd(o + 0, we * v.x);
    atomicAdd(o + 1, we * v.y);
    atomicAdd(o + 2, we * v.z);
    atomicAdd(o + 3, we * v.w);
}

// ---------------- WMMA f32 GEMM: out = relu(A1@B1 [+ A2@B2] + bias) [+ resid] ---------
// One wave per 16x16 output tile; 4 waves (4 M-tiles) per 128-thread block.
// V_WMMA_F32_16X16X4_F32 operand layout (ISA 7.12.2):
//   A (16x4): lane (l&15)=row, VGPR j holds K = 2*(l>>4)+j
//   B (4x16): lane (l&15)=col, VGPR j holds K = 2*(l>>4)+j
//   C (16x16): VGPR v holds row (l<16 ? v : v+8), col = l&15
// B comes from the pair-packed buffers (k_pack_w), so the inner loop is branch-free:
// per K-step just unconditional b64 loads + v_wmma.

template <bool DUAL, bool RESID>
__global__ void k_gemm_wmma(const float* __restrict__ A1, const float* __restrict__ P1,
                            const float* __restrict__ A2, const float* __restrict__ P2,
                            const float* __restrict__ bias,
                            const float* __restrict__ resid,
                            float* __restrict__ out,
                            int M, int K, int lda, int Nout, int Np, int ldo) {
    const int lane  = threadIdx.x & 31;
    const int wv    = threadIdx.x >> 5;
    const int mtile = blockIdx.y * 4 + wv;
    const int m0    = mtile * 16;
    if (m0 >= M) return;
    const int n0   = blockIdx.x * 16;
    const int rrow = lane & 15;              // also the output column sub-index
    const int hi   = lane >> 4;
    const int kb   = hi * 2;                 // K sub-offset for this half-wave
    const int ncol = n0 + rrow;

    v8f c = {};
    const float* a1p = A1 + (size_t)(m0 + rrow) * lda + kb;
    const float* p1  = P1 + ((size_t)hi * Np + ncol) * 2;   // += k*Np per step (floats)
    const float* a2p = DUAL ? (A2 + (size_t)(m0 + rrow) * lda + kb) : nullptr;
    const float* p2  = DUAL ? (P2 + ((size_t)hi * Np + ncol) * 2) : nullptr;

    for (int k = 0; k < K; k += 4) {
        v2f a = *(const v2f*)(a1p + k);
        v2f b = *(const v2f*)(p1 + (size_t)k * Np);
        c = __builtin_amdgcn_wmma_f32_16x16x4_f32(false, a, false, b,
                                                  (short)0, c, false, false);
        if (DUAL) {
            v2f a2 = *(const v2f*)(a2p + k);
            v2f b2 = *(const v2f*)(p2 + (size_t)k * Np);
            c = __builtin_amdgcn_wmma_f32_16x16x4_f32(false, a2, false, b2,
                                                      (short)0, c, false, false);
        }
    }

    if (ncol >= Nout) return;                // pad columns: never written, never read
    float bb = bias[ncol];
    for (int v = 0; v < 8; ++v) {
        int m = m0 + (hi ? (v + 8) : v);
        float val = fmaxf(c[v] + bb, 0.0f);  // all uses of this kernel apply ReLU
        if (RESID) val += resid[(size_t)m * ldo + ncol];
        out[(size_t)m * ldo + ncol] = val;
    }
}

// ---------------- link-prediction loss: one wave per edge, float4 lanes ----------------

__global__ void k_edge_loss(const int* __restrict__ ei, const float* __restrict__ z,
                            int E, int ld, int is_neg, float* __restrict__ acc) {
    int e    = blockIdx.x * (blockDim.x >> 5) + (threadIdx.x >> 5);
    int lane = threadIdx.x & 31;
    if (e >= E) return;
    const float* za = z + (size_t)ei[e]     * ld;
    const float* zb = z + (size_t)ei[E + e] * ld;
    int   c4  = (lane < 25 ? lane : 24) * 4;          // F=100 -> 25 float4 chunks
    float msk = (lane < 25) ? 1.0f : 0.0f;
    const float4 a = *(const float4*)(za + c4);
    const float4 b = *(const float4*)(zb + c4);
    float s = msk * (a.x * b.x + a.y * b.y + a.z * b.z + a.w * b.w);
    for (int off = 16; off > 0; off >>= 1) s += __shfl_xor(s, off, 32);
    if (lane == 0) {
        float sig  = 1.0f / (1.0f + expf(-s));
        float term = is_neg ? logf(1.0f - sig + 1e-15f) : logf(sig + 1e-15f);
        atomicAdd(acc, term);
    }
}

// ---------------- final Cheb layer, OUT=1: one wave per node ----------------

__global__ void k_out(const float* __restrict__ xm, const float* __restrict__ txm,
                      const float* __restrict__ W3a, const float* __restrict__ W3b,
                      const float* __restrict__ b3, float* __restrict__ out,
                      int M, int ld) {
    int i    = blockIdx.x * (blockDim.x >> 5) + (threadIdx.x >> 5);
    int lane = threadIdx.x & 31;
    if (i >= M) return;
    int   c4  = (lane < 25 ? lane : 24) * 4;
    float msk = (lane < 25) ? 1.0f : 0.0f;
    const float4 a  = *(const float4*)(xm  + (size_t)i * ld + c4);
    const float4 t  = *(const float4*)(txm + (size_t)i * ld + c4);
    const float4 wa = *(const float4*)(W3a + c4);
    const float4 wb = *(const float4*)(W3b + c4);
    float s = msk * (a.x * wa.x + a.y * wa.y + a.z * wa.z + a.w * wa.w +
                     t.x * wb.x + t.y * wb.y + t.z * wb.z + t.w * wb.w);
    for (int off = 16; off > 0; off >>= 1) s += __shfl_xor(s, off, 32);
    if (lane == 0) out[i] = s + b3[0];
}

__global__ void k_finalize(const float* __restrict__ acc,
                           const float* __restrict__ c1, const float* __restrict__ c2,
                           float* __restrict__ tail, float invE) {
    tail[0] = -(acc[0] + acc[1]) * invE;   // pos_loss + neg_loss
    tail[1] = c1[0];
    tail[2] = c2[0];
}

// ---------------- driver ----------------

extern "C" void kernel_launch(void* const* d_in, const int* in_sizes, int n_in,
                              void* d_out, int out_size, void* d_ws, size_t ws_size,
                              hipStream_t stream) {
    (void)in_sizes; (void)n_in; (void)out_size; (void)ws_size;

    const float* x      = (const float*)d_in[0];
    const int*   ei     = (const int*)  d_in[1];
    const int*   nei    = (const int*)  d_in[2];
    const float* W1a    = (const float*)d_in[3];
    const float* W1b    = (const float*)d_in[4];
    const float* b1     = (const float*)d_in[5];
    const float* W2a    = (const float*)d_in[6];
    const float* W2b    = (const float*)d_in[7];
    const float* b2     = (const float*)d_in[8];
    const float* W3a    = (const float*)d_in[9];
    const float* W3b    = (const float*)d_in[10];
    const float* b3     = (const float*)d_in[11];
    const float* lin1_W = (const float*)d_in[12];
    const float* lin1_b = (const float*)d_in[13];
    const float* lin2_W = (const float*)d_in[14];
    const float* lin2_b = (const float*)d_in[15];
    const float* c1     = (const float*)d_in[16];
    const float* c2     = (const float*)d_in[17];

    const int N = 50000, E = 800000;
    const int LD1 = 304;   // H1=300 padded row stride (19 n-tiles)
    const int LD2 = 112;   // H2=100 padded row stride (7 n-tiles)

    float* ws    = (float*)d_ws;
    float* deg   = ws;  ws += N;
    float* dinv  = ws;  ws += N;
    float* w     = ws;  ws += E;
    float* tx1x  = ws;  ws += (size_t)N * 64;
    float* h     = ws;  ws += (size_t)N * LD1;
    float* tx1h  = ws;  ws += (size_t)N * LD1;
    float* x1    = ws;  ws += (size_t)N * LD2;
    float* xm    = ws;  ws += (size_t)N * LD2;
    float* z     = ws;  ws += (size_t)N * LD2;
    float* tx1xm = ws;  ws += (size_t)N * LD2;
    float* pW1a  = ws;  ws += (size_t)64  * LD1;
    float* pW1b  = ws;  ws += (size_t)64  * LD1;
    float* pW2a  = ws;  ws += (size_t)300 * LD2;
    float* pW2b  = ws;  ws += (size_t)300 * LD2;
    float* pL1   = ws;  ws += (size_t)64  * LD2;
    float* pL2   = ws;  ws += (size_t)64  * LD2;
    float* acc   = ws;  ws += 2;

    // atomic-scatter targets must start at zero every call
    hipMemsetAsync(deg,   0, (size_t)N * sizeof(float), stream);
    hipMemsetAsync(tx1x,  0, (size_t)N * 64  * sizeof(float), stream);
    hipMemsetAsync(tx1h,  0, (size_t)N * LD1 * sizeof(float), stream);
    hipMemsetAsync(tx1xm, 0, (size_t)N * LD2 * sizeof(float), stream);
    hipMemsetAsync(acc,   0, 2 * sizeof(float), stream);

    dim3 b256(256), b128(128);
    const int mblk = (N / 16 + 3) / 4;   // 4 M-tiles per 128-thread block

    // normalization
    k_deg <<<dim3((E + 255) / 256), b256, 0, stream>>>(ei, deg, E);
    k_dinv<<<dim3((N + 255) / 256), b256, 0, stream>>>(deg, dinv, N);
    k_w   <<<dim3((E + 255) / 256), b256, 0, stream>>>(ei, dinv, w, E);

    // weight packing (pair-interleaved + zero-padded columns)
    k_pack_w<<<dim3((64  * LD1 + 255) / 256), b256, 0, stream>>>(W1a,    pW1a, 64,  300, LD1);
    k_pack_w<<<dim3((64  * LD1 + 255) / 256), b256, 0, stream>>>(W1b,    pW1b, 64,  300, LD1);
    k_pack_w<<<dim3((300 * LD2 + 255) / 256), b256, 0, stream>>>(W2a,    pW2a, 300, 100, LD2);
    k_pack_w<<<dim3((300 * LD2 + 255) / 256), b256, 0, stream>>>(W2b,    pW2b, 300, 100, LD2);
    k_pack_w<<<dim3((64  * LD2 + 255) / 256), b256, 0, stream>>>(lin1_W, pL1,  64,  100, LD2);
    k_pack_w<<<dim3((64  * LD2 + 255) / 256), b256, 0, stream>>>(lin2_W, pL2,  64,  100, LD2);

    // tx1_x = L_hat @ x   [N,64]
    k_spmm<<<dim3((E * 16 + 255) / 256), b256, 0, stream>>>(ei, w, x, tx1x, E, 16, 64);

    // h = relu(x@W1a + tx1_x@W1b + b1)   [N,300] (ld 304)
    k_gemm_wmma<true, false><<<dim3(LD1 / 16, mblk), b128, 0, stream>>>(
        x, pW1a, tx1x, pW1b, b1, nullptr, h, N, 64, 64, 300, LD1, LD1);

    // tx1_h = L_hat @ h   [N,300]
    k_spmm<<<dim3((E * 75 + 255) / 256), b256, 0, stream>>>(ei, w, h, tx1h, E, 75, LD1);

    // x1 = relu(h@W2a + tx1_h@W2b + b2)   [N,100] (ld 112)
    k_gemm_wmma<true, false><<<dim3(LD2 / 16, mblk), b128, 0, stream>>>(
        h, pW2a, tx1h, pW2b, b2, nullptr, x1, N, 300, LD1, 100, LD2, LD2);

    // xm = x1 + relu(x@lin1_W + lin1_b) ; z = x1 + relu(x@lin2_W + lin2_b)
    k_gemm_wmma<false, true><<<dim3(LD2 / 16, mblk), b128, 0, stream>>>(
        x, pL1, nullptr, nullptr, lin1_b, x1, xm, N, 64, 64, 100, LD2, LD2);
    k_gemm_wmma<false, true><<<dim3(LD2 / 16, mblk), b128, 0, stream>>>(
        x, pL2, nullptr, nullptr, lin2_b, x1, z, N, 64, 64, 100, LD2, LD2);

    // reconstruction loss: one wave per edge, 4 waves per block
    k_edge_loss<<<dim3((E + 3) / 4), b128, 0, stream>>>(ei,  z, E, LD2, 0, acc + 0);
    k_edge_loss<<<dim3((E + 3) / 4), b128, 0, stream>>>(nei, z, E, LD2, 1, acc + 1);

    // tx1_xm = L_hat @ xm   [N,100]
    k_spmm<<<dim3((E * 25 + 255) / 256), b256, 0, stream>>>(ei, w, xm, tx1xm, E, 25, LD2);

    // out = xm@W3a + tx1_xm@W3b + b3   [N,1] : one wave per node
    float* out = (float*)d_out;
    k_out<<<dim3((N + 3) / 4), b128, 0, stream>>>(xm, tx1xm, W3a, W3b, b3, out, N, LD2);

    // tail: [r_loss, c1, c2]
    k_finalize<<<dim3(1), dim3(1), 0, stream>>>(acc, c1, c2, out + N, 1.0f / (float)E);
}